// DigitCaps_19464791786185
// MI455X (gfx1250) — compile-verified
//
#include <hip/hip_runtime.h>
#include <hip/hip_bf16.h>

typedef __attribute__((ext_vector_type(2))) float v2f;
typedef __attribute__((ext_vector_type(8))) float v8f;

#define BS_  512
#define P_   1152
#define C_   10
#define D_   16
#define E_   8
#define K_   (P_*E_)   // 9216
#define N_   (C_*D_)   // 160

// One block = 4 waves, each wave owns a K-chunk of 2304 (288 p-groups of 8).
// blockIdx.x = nt (N tile, 0..9), blockIdx.y = bt (M tile over batch, 0..31).
__global__ __launch_bounds__(128)
void digitcaps_wmma_f32(const float* __restrict__ x,
                        const float* __restrict__ W,
                        float* __restrict__ out)
{
    const int lane = threadIdx.x & 31;
    const int wave = threadIdx.x >> 5;   // 0..3 : K-split
    const int m    = lane & 15;          // row within 16-row A tile / col within B tile
    const int half = lane >> 4;          // lane-half selects K+0 vs K+2 in fragments
    const int nt   = blockIdx.x;         // 0..9
    const int bt   = blockIdx.y;         // 0..31

    // A fragment source: x[bt*16 + m][k], K contiguous (k = p*8 + e).
    const float* __restrict__ xA =
        x + (size_t)(bt * 16 + m) * K_ + (size_t)wave * 2304 + 2 * half;

    // B fragment source: W[p, c=nt, d=m, e]; flat = p*1280 + nt*128 + m*8 + e.
    const float* __restrict__ wB =
        W + (size_t)(wave * 288) * 1280 + (size_t)nt * 128 + m * 8 + 2 * half;

    v8f acc = {0.f, 0.f, 0.f, 0.f, 0.f, 0.f, 0.f, 0.f};

    // 288 p-groups; each p-group = 8 K values = two 16x16x4 f32 WMMAs.
    #pragma unroll 4
    for (int it = 0; it < 288; ++it) {
        v2f a0 = *(const v2f*)(xA);        // K = p*8   + {0,1}/{2,3} by lane-half
        v2f b0 = *(const v2f*)(wB);
        v2f a1 = *(const v2f*)(xA + 4);    // K = p*8+4 + {0,1}/{2,3}
        v2f b1 = *(const v2f*)(wB + 4);

        // 8-arg pattern: (neg_a, A, neg_b, B, c_mod, C, reuse_a, reuse_b)
        acc = __builtin_amdgcn_wmma_f32_16x16x4_f32(
                  false, a0, false, b0, (short)0, acc, false, false);
        acc = __builtin_amdgcn_wmma_f32_16x16x4_f32(
                  false, a1, false, b1, (short)0, acc, false, false);

        xA += 8;      // next p-group in x (K contiguous)
        wB += 1280;   // next p-group in W (p stride = C*D*E)
    }

    // Cross-wave K reduction through LDS (pad inner dim to 9 -> conflict-free).
    __shared__ float red[4][32][9];
    #pragma unroll
    for (int i = 0; i < 8; ++i) red[wave][lane][i] = acc[i];
    __syncthreads();

    if (wave == 0) {
        #pragma unroll
        for (int i = 0; i < 8; ++i) {
            float s = red[0][lane][i] + red[1][lane][i]
                    + red[2][lane][i] + red[3][lane][i];
            s *= (1.0f / (float)P_);                 // mean over P (uniform routing)
            float v = s * fabsf(s) / (1.0f + s * s); // degenerate squash
            // C/D fragment: VGPR i holds M = i + 8*half, N = m.
            out[(size_t)(bt * 16 + i + 8 * half) * N_ + nt * 16 + m] = v;
        }
    }
}

extern "C" void kernel_launch(void* const* d_in, const int* in_sizes, int n_in,
                              void* d_out, int out_size, void* d_ws, size_t ws_size,
                              hipStream_t stream)
{
    const float* x = (const float*)d_in[0];   // (512, 1152, 8)
    const float* W = (const float*)d_in[1];   // (1, 1152, 10, 16, 8)
    float* out = (float*)d_out;               // (512, 10, 16, 1)

    dim3 grid(C_ /*nt=10*/, BS_ / 16 /*bt=32*/);
    digitcaps_wmma_f32<<<grid, 128, 0, stream>>>(x, W, out);
}